// PrimeResonantFilter_79207786872866
// MI455X (gfx1250) — compile-verified
//
#include <hip/hip_runtime.h>
#include <math.h>

typedef float v2f __attribute__((ext_vector_type(2)));
typedef float v4f __attribute__((ext_vector_type(4)));
typedef float v8f __attribute__((ext_vector_type(8)));

#define PRF_PI 3.14159265358979323846f

// ---------------------------------------------------------------------------
// Prep kernel: collapse the 24-prime Gaussian filter into 4 real circulant
// taps c[0..3].  out[j] = sum_m c[(j-m)&3] * x[m].   Runs once, one lane.
// ---------------------------------------------------------------------------
__global__ void prf_prep_kernel(const float* __restrict__ amp,
                                const float* __restrict__ phase,
                                float* __restrict__ c_out) {
    if (threadIdx.x != 0 || blockIdx.x != 0) return;
    const float primes[24] = {2,3,5,7,11,13,17,19,23,29,31,37,
                              41,43,47,53,59,61,67,71,73,79,83,89};
    float re[4] = {0.f,0.f,0.f,0.f};
    float im[4] = {0.f,0.f,0.f,0.f};
    for (int p = 0; p < 24; ++p) {
        float pf = primes[p] * (PRF_PI / 89.0f);
        float a  = amp[p];
        float cp = cosf(phase[p]);
        float sp = sinf(phase[p]);
        for (int k = 0; k < 4; ++k) {
            float fi = (float)k * (PRF_PI / 4.0f);
            float t  = (fi - pf) * 10.0f;          // sigma = 0.1
            float g  = expf(-0.5f * t * t);
            re[k] += a * g * cp;
            im[k] += a * g * sp;
        }
    }
    float mx = 0.f;
    for (int k = 0; k < 4; ++k)
        mx = fmaxf(mx, sqrtf(re[k]*re[k] + im[k]*im[k]));
    float inv = 1.0f / (mx + 1e-8f);
    for (int k = 0; k < 4; ++k) { re[k] *= inv; im[k] *= inv; }
    c_out[0] = 0.25f * (re[0] + re[1] + re[2] + re[3]);
    c_out[1] = 0.25f * (re[0] - im[1] - re[2] + im[3]);
    c_out[2] = 0.25f * (re[0] - re[1] + re[2] - re[3]);
    c_out[3] = 0.25f * (re[0] + im[1] - re[2] - im[3]);
}

// ---------------------------------------------------------------------------
// Main kernel: V_WMMA_F32_16X16X4_F32, 4 tiles (64 groups, 1 KiB) per wave
// iteration for memory-level parallelism.
//   A (16x4): rows 0..3 = circulant M[j][m] = c[(j-m)&3], rows 4..15 = 0
//     lane M : A[M,0],A[M,1] ; lane M+16 : A[M,2],A[M,3]
//   B (4x16): column N = group N's 4 floats
//     lane N : B[0,N],B[1,N] ; lane N+16 : B[2,N],B[3,N]
//     -> lane L loads b64 at tile + 16*(L&15) + 8*(L>>4): coalesced 256B/wave
//   D (16x16): lane N holds out_N[0..3] in VGPRs 0..3 -> float4 store, lanes 0-15
// ---------------------------------------------------------------------------
__global__ void prf_apply_kernel(const float* __restrict__ in,
                                 float* __restrict__ out,
                                 const float* __restrict__ c,
                                 long long num_tiles,      // tiles of 16 groups
                                 long long num_groups) {   // total groups of 4
    const int lane = threadIdx.x & 31;
    const int row  = lane & 15;
    const int hi   = lane >> 4;        // 0: K=0,1   1: K=2,3
    const int col0 = hi * 2;

    // Per-lane A fragment (2 VGPRs, constant for whole kernel).
    v2f a;
    a.x = (row < 4) ? c[(row - col0    ) & 3] : 0.0f;
    a.y = (row < 4) ? c[(row - col0 - 1) & 3] : 0.0f;

    const long long wave   = (long long)blockIdx.x * (blockDim.x >> 5)
                           + (long long)(threadIdx.x >> 5);
    const long long nwaves = (long long)gridDim.x * (blockDim.x >> 5);
    const int boff = 4 * row + 2 * hi;   // float offset of this lane's b64 load
    const int soff = 4 * row;            // float offset of this lane's b128 store

    // ---- quad-tile main loop: 4 independent loads -> 4 WMMAs -> 4 stores ----
    const long long num_quads = num_tiles >> 2;
    for (long long q = wave; q < num_quads; q += nwaves) {
        const float* tin  = in  + q * 256;
        float*       tout = out + q * 256;
        v2f b0 = __builtin_nontemporal_load((const v2f*)(tin       + boff));
        v2f b1 = __builtin_nontemporal_load((const v2f*)(tin +  64 + boff));
        v2f b2 = __builtin_nontemporal_load((const v2f*)(tin + 128 + boff));
        v2f b3 = __builtin_nontemporal_load((const v2f*)(tin + 192 + boff));
        v8f z = {};
        v8f d0 = __builtin_amdgcn_wmma_f32_16x16x4_f32(false, a, false, b0, (short)0, z, false, false);
        v8f d1 = __builtin_amdgcn_wmma_f32_16x16x4_f32(false, a, false, b1, (short)0, z, false, false);
        v8f d2 = __builtin_amdgcn_wmma_f32_16x16x4_f32(false, a, false, b2, (short)0, z, false, false);
        v8f d3 = __builtin_amdgcn_wmma_f32_16x16x4_f32(false, a, false, b3, (short)0, z, false, false);
        if (hi == 0) {                    // single exec toggle for all 4 stores
            v4f r0; r0.x = d0[0]; r0.y = d0[1]; r0.z = d0[2]; r0.w = d0[3];
            v4f r1; r1.x = d1[0]; r1.y = d1[1]; r1.z = d1[2]; r1.w = d1[3];
            v4f r2; r2.x = d2[0]; r2.y = d2[1]; r2.z = d2[2]; r2.w = d2[3];
            v4f r3; r3.x = d3[0]; r3.y = d3[1]; r3.z = d3[2]; r3.w = d3[3];
            __builtin_nontemporal_store(r0, (v4f*)(tout       + soff));
            __builtin_nontemporal_store(r1, (v4f*)(tout +  64 + soff));
            __builtin_nontemporal_store(r2, (v4f*)(tout + 128 + soff));
            __builtin_nontemporal_store(r3, (v4f*)(tout + 192 + soff));
        }
    }

    // ---- leftover tiles (< 4): single-tile WMMA loop ----
    for (long long t = num_quads * 4 + wave; t < num_tiles; t += nwaves) {
        const float* tin = in + t * 64;
        v2f b = __builtin_nontemporal_load((const v2f*)(tin + boff));
        v8f z = {};
        v8f d = __builtin_amdgcn_wmma_f32_16x16x4_f32(false, a, false, b, (short)0, z, false, false);
        if (hi == 0) {
            v4f r; r.x = d[0]; r.y = d[1]; r.z = d[2]; r.w = d[3];
            __builtin_nontemporal_store(r, (v4f*)(out + t * 64 + soff));
        }
    }

    // ---- scalar tail for group counts not divisible by 16 ----
    const long long tid      = (long long)blockIdx.x * blockDim.x + threadIdx.x;
    const long long nthreads = (long long)gridDim.x * blockDim.x;
    const float c0 = c[0], c1 = c[1], c2 = c[2], c3 = c[3];
    for (long long g = num_tiles * 16 + tid; g < num_groups; g += nthreads) {
        const float x0 = in[4*g+0], x1 = in[4*g+1], x2 = in[4*g+2], x3 = in[4*g+3];
        out[4*g+0] = c0*x0 + c3*x1 + c2*x2 + c1*x3;
        out[4*g+1] = c1*x0 + c0*x1 + c3*x2 + c2*x3;
        out[4*g+2] = c2*x0 + c1*x1 + c0*x2 + c3*x3;
        out[4*g+3] = c3*x0 + c2*x1 + c1*x2 + c0*x3;
    }
}

// ---------------------------------------------------------------------------
extern "C" void kernel_launch(void* const* d_in, const int* in_sizes, int n_in,
                              void* d_out, int out_size, void* d_ws, size_t ws_size,
                              hipStream_t stream) {
    const float* q   = (const float*)d_in[0];   // quantum_state  [4,2048,1024,4]
    const float* amp = (const float*)d_in[1];   // resonance_amplitude [24]
    const float* ph  = (const float*)d_in[2];   // resonance_phase     [24]
    float* out = (float*)d_out;
    float* c   = (float*)d_ws;                  // 4 floats of scratch

    const long long n          = (long long)in_sizes[0]; // total fp32 elements
    const long long num_groups = n >> 2;                 // quaternion groups
    const long long num_tiles  = num_groups >> 4;        // 16 groups per WMMA

    prf_prep_kernel<<<1, 32, 0, stream>>>(amp, ph, c);

    const int block = 256;                               // 8 wave32 per block
    const int waves_per_block = block / 32;
    const long long num_quads = num_tiles >> 2;
    long long want = (num_quads + waves_per_block - 1) / waves_per_block;
    if (want < 1) want = 1;
    const int grid = (int)((want < 4096) ? want : 4096);

    prf_apply_kernel<<<grid, block, 0, stream>>>(q, out, c, num_tiles, num_groups);
}